// YoloLayer_2388001817145
// MI455X (gfx1250) — compile-verified
//
#include <hip/hip_runtime.h>

// YOLO head layer for MI455X (gfx1250).
// Bandwidth-bound stream (377 MB @ 23.3 TB/s ~= 16 us floor). Strategy:
//  - block = 256 threads owns 4 pixels = 1020 floats = 255 float4 quads
//    (group base is a multiple of 255 => channel = in-group offset mod 255)
//  - constexpr LUT permutes quads so the 21 "special" quads (xy/wh channels)
//    all land in wave 7; waves 0-6 are pure sigmoid with zero index math and
//    zero divergence (b128 NT load -> 4x v_exp_f32/v_rcp_f32 -> b128 NT store)
//  - pixel coords (%76) computed from uniform blockIdx => SALU magic divides
//  - gfx1250 paths: th:TH_LOAD_NT / th:TH_STORE_NT b128, global_prefetch_b8

typedef __attribute__((ext_vector_type(4))) float v4f;

namespace {

constexpr float kLog2e = 1.4426950408889634f;
constexpr float kInvW  = 1.0f / 76.0f;

__device__ __forceinline__ float fast_sigmoid(float v) {
  float e = __builtin_amdgcn_exp2f(-v * kLog2e);     // 2^(-v*log2e) = exp(-v)
  return __builtin_amdgcn_rcpf(1.0f + e);
}

// One element, general path. c = channel 0..254, d = pixel offset in group.
// w_b/h_b are the block's base pixel coords (uniform, SALU-computed).
__device__ __forceinline__ float transform_general(float val, unsigned c,
                                                   unsigned d, unsigned w_b,
                                                   unsigned h_b) {
  unsigned a  = (c >= 170u) ? 2u : ((c >= 85u) ? 1u : 0u);   // anchor
  unsigned ch = c - a * 85u;                                 // 0..84
  float r;
  if (ch >= 4u) {
    r = fast_sigmoid(val);                                   // obj + classes
  } else if (ch >= 2u) {                                     // wh
    float aw = (a == 0u) ? (12.0f / 608.0f)
                         : ((a == 1u) ? (19.0f / 608.0f) : (40.0f / 608.0f));
    float ah = (a == 0u) ? (16.0f / 608.0f)
                         : ((a == 1u) ? (36.0f / 608.0f) : (28.0f / 608.0f));
    float an = (ch == 2u) ? aw : ah;
    r = __builtin_amdgcn_exp2f(val * kLog2e) * an;
  } else {                                                   // xy
    unsigned wd   = w_b + d;                                 // d <= 3
    bool     wrap = wd >= 76u;
    unsigned w    = wrap ? (wd - 76u) : wd;
    unsigned h    = wrap ? ((h_b + 1u >= 76u) ? 0u : (h_b + 1u)) : h_b;
    float g = (ch == 0u) ? (float)w : (float)h;
    // ((s-0.5)*1.2+0.5 + g)/76 == (1.2*s - 0.1 + g)/76
    r = fmaf(fast_sigmoid(val), 1.2f, g - 0.1f) * kInvW;
  }
  return r;
}

struct Lut { unsigned v[255]; };

// Entry: bits[7:0]=quad index q, [15:8]=c0=(4q)%255, [17:16]=poff=(4q)/255,
// [18]=special. Specials (21 of them) placed at slots 224..244 => wave 7.
constexpr Lut make_lut() {
  Lut L{};
  int pure_lo = 0, pure_hi = 245, sp = 224;
  for (int q = 0; q < 255; ++q) {
    bool special = false;
    for (int j = 0; j < 4; ++j)
      if ((unsigned)(4 * q + j) % 85u < 4u) special = true;
    unsigned c0   = (unsigned)(4 * q) % 255u;
    unsigned poff = (unsigned)(4 * q) / 255u;
    unsigned e = (unsigned)q | (c0 << 8) | (poff << 16)
               | (special ? (1u << 18) : 0u);
    int pos;
    if (special)            pos = sp++;       // 224..244 (exactly 21)
    else if (pure_lo < 224) pos = pure_lo++;  // 0..223
    else                    pos = pure_hi++;  // 245..254
    L.v[pos] = e;
  }
  return L;
}

} // namespace

__constant__ Lut kLut = make_lut();

__global__ __launch_bounds__(256) void yolo_head_kernel(
    const float* __restrict__ x, float* __restrict__ out, int n) {
  const unsigned t = threadIdx.x;
  if (t >= 255u) return;                       // 255 quads per block

  const unsigned entry = kLut.v[t];
  const unsigned q     = entry & 0xFFu;
  const unsigned base  = blockIdx.x * 1020u + q * 4u;   // flat element index
  if (base >= (unsigned)n) return;

  // Streaming read-ahead, ~64KB (16 blocks) ahead. Speculative, OOB dropped.
  __builtin_prefetch(x + base + 16320u, /*rw=*/0, /*locality=*/2);

  if (base + 4u <= (unsigned)n) {
    v4f v = __builtin_nontemporal_load((const v4f*)(x + base));
    v4f r;
    if ((entry & (1u << 18)) == 0u) {
      // Pure path (waves 0-6 entirely): 4 sigmoids, no index math.
      r.x = fast_sigmoid(v.x);
      r.y = fast_sigmoid(v.y);
      r.z = fast_sigmoid(v.z);
      r.w = fast_sigmoid(v.w);
    } else {
      // Special path: confined to wave 7 by the LUT permutation.
      const unsigned c0    = (entry >> 8) & 0xFFu;
      const unsigned poff  = (entry >> 16) & 3u;
      const unsigned p_blk = blockIdx.x * 4u;           // uniform => SALU:
      const unsigned w_b   = p_blk % 76u;
      const unsigned h_b   = (p_blk / 76u) % 76u;
      float vv[4] = {v.x, v.y, v.z, v.w};
      float rr[4];
#pragma unroll
      for (unsigned j = 0; j < 4u; ++j) {
        unsigned c       = c0 + j;
        unsigned wrapped = (c >= 255u) ? 1u : 0u;       // crosses pixel
        c -= wrapped * 255u;
        rr[j] = transform_general(vv[j], c, poff + wrapped, w_b, h_b);
      }
      r.x = rr[0]; r.y = rr[1]; r.z = rr[2]; r.w = rr[3];
    }
    __builtin_nontemporal_store(r, (v4f*)(out + base));
  } else {
    // Ragged tail (never hit for this shape: n % 1020 == 0). Scalar, general.
    const unsigned c0    = (entry >> 8) & 0xFFu;
    const unsigned poff  = (entry >> 16) & 3u;
    const unsigned p_blk = blockIdx.x * 4u;
    const unsigned w_b   = p_blk % 76u;
    const unsigned h_b   = (p_blk / 76u) % 76u;
    for (unsigned j = 0; base + j < (unsigned)n; ++j) {
      unsigned c       = c0 + j;
      unsigned wrapped = (c >= 255u) ? 1u : 0u;
      c -= wrapped * 255u;
      out[base + j] = transform_general(x[base + j], c, poff + wrapped, w_b, h_b);
    }
  }
}

extern "C" void kernel_launch(void* const* d_in, const int* in_sizes, int n_in,
                              void* d_out, int out_size, void* d_ws, size_t ws_size,
                              hipStream_t stream) {
  (void)n_in; (void)d_ws; (void)ws_size; (void)out_size;
  const float* x = (const float*)d_in[0];
  float* out     = (float*)d_out;
  const int n    = in_sizes[0];               // 32*76*76*255 = 47,132,160

  const int block = 256;                      // 8 wave32; wave 7 = special wave
  const int grid  = (n + 1019) / 1020;        // one block per 4 pixels (exact)

  yolo_head_kernel<<<grid, block, 0, stream>>>(x, out, n);
}